// ACmix_83459804496116
// MI455X (gfx1250) — compile-verified
//
#include <hip/hip_runtime.h>
#include <hip/hip_bf16.h>

// ---------------- problem constants ----------------
#define C_    256
#define HEAD_ 4
#define HD_   64
#define B_    2
#define H_    48
#define W_    48
#define N_    (H_*W_)        // 2304 pixels per image
#define PLANE_ (C_*N_)       // 589824 floats per batch image
#define QSIZE_ (B_*PLANE_)   // 1179648 floats per q/k/v buffer

typedef __attribute__((ext_vector_type(2))) float v2f;
typedef __attribute__((ext_vector_type(8))) float v8f;

// reflect padding (jnp 'reflect': p[-1]=p[1], p[48]=p[46]); max overshoot is 3
__device__ __forceinline__ int refl(int i) {
  return i < 0 ? -i : (i >= H_ ? 2*H_ - 2 - i : i);
}

// per-lane async copy global -> LDS (ASYNCcnt-tracked, CDNA5)
__device__ __forceinline__ void async_g2l_b32(unsigned laddr, const float* gp) {
  asm volatile("global_load_async_to_lds_b32 %0, %1, off"
               :: "v"(laddr), "v"(gp) : "memory");
}
__device__ __forceinline__ void async_g2l_b128(unsigned laddr, const float* gp) {
  asm volatile("global_load_async_to_lds_b128 %0, %1, off"
               :: "v"(laddr), "v"(gp) : "memory");
}
__device__ __forceinline__ void wait_async0() {
  asm volatile("s_wait_asynccnt 0x0" ::: "memory");
}

// ---------------------------------------------------------------------------
// Kernel 1: q/k/v projections as f32 WMMA GEMM.
//   D[o,n] = sum_c W[o,c] * X[c,n] ; then (D + bias) * scale.
//   grid.y in {0,1,2} selects q/k/v.  128 threads = 4 waves; all 4 waves share
//   one 16-row W tile (16 KB) staged in LDS once per block with
//   GLOBAL_LOAD_ASYNC_TO_LDS_B128; each wave owns one 16x16 output tile.
// WMMA f32 16x16x4 lane mapping (ISA 7.12.2):
//   A (16x4):  lane<16 -> M=lane, {v0,v1}={K0,K1}; lane>=16 -> {K2,K3}
//   B (4x16):  lane<16 -> N=lane, {v0,v1}={K0,K1}; lane>=16 -> {K2,K3}
//   C/D:       vgpr i  -> M = m0 + i + 8*(lane>=16), N = n0 + (lane&15)
// LDS A-tile row stride 260 (== 4 mod 64 banks): lane r hits banks {4r,4r+1}
// on the ds_load_b64 -> conflict-free.
// ---------------------------------------------------------------------------
#define SW_STRIDE 260

__global__ __launch_bounds__(128) void qkv_wmma_kernel(
    const float* __restrict__ x,
    const float* __restrict__ Wq, const float* __restrict__ bq,
    const float* __restrict__ Wk, const float* __restrict__ bk,
    const float* __restrict__ Wv, const float* __restrict__ bv,
    float* __restrict__ ws)
{
  __shared__ float sW[16 * SW_STRIDE];

  const int which = blockIdx.y;
  const float* Wm = which == 0 ? Wq : (which == 1 ? Wk : Wv);
  const float* bm = which == 0 ? bq : (which == 1 ? bk : bv);
  const float scale = which == 0 ? 0.125f : 1.0f;   // q *= HD^-0.5
  float* dst = ws + (size_t)which * QSIZE_;

  const int lane = threadIdx.x & 31;
  const int wave = threadIdx.x >> 5;
  const int mt = blockIdx.x & 15;            // 16 m-tiles (M = 256)
  const int ng = blockIdx.x >> 4;            // 72 n-groups
  const int nt = ng * 4 + wave;              // 288 n-tiles (N = 4608)
  const int m0 = mt * 16;
  const int n0 = nt * 16;
  const int b  = n0 / N_;                    // n-tiles never straddle batches
  const int row = lane & 15;
  const int hi  = lane >> 4;
  const int p   = (n0 % N_) + row;

  // ---- stage W[m0:m0+16, 0:256] into LDS: 1024 x 16B async chunks ----
  {
    const float* gW = Wm + (size_t)m0 * C_;
    #pragma unroll
    for (int it = 0; it < 8; ++it) {
      int cid  = it * 128 + threadIdx.x;     // chunk id, 64 chunks per row
      int rowi = cid >> 6;
      int ci   = cid & 63;
      unsigned laddr = (unsigned)(size_t)&sW[rowi * SW_STRIDE + ci * 4];
      async_g2l_b128(laddr, gW + rowi * C_ + ci * 4);
    }
    wait_async0();
    __syncthreads();
  }

  const float* Arow = sW + row * SW_STRIDE + 2 * hi;           // LDS
  const float* Bcol = x + (size_t)b * PLANE_ + (size_t)(2 * hi) * N_ + p;

  v8f acc = {};
  #pragma unroll 8
  for (int k0 = 0; k0 < C_; k0 += 4) {
    v2f a = *(const v2f*)(Arow + k0);        // ds_load_b64, conflict-free
    v2f bb;
    bb.x = Bcol[(size_t)k0 * N_];
    bb.y = Bcol[(size_t)k0 * N_ + N_];
    acc = __builtin_amdgcn_wmma_f32_16x16x4_f32(false, a, false, bb,
                                                (short)0, acc, false, false);
  }

  float* outb = dst + (size_t)b * PLANE_ + p;
  #pragma unroll
  for (int i = 0; i < 8; ++i) {
    int m = m0 + i + 8 * hi;
    outb[(size_t)m * N_] = (acc[i] + bm[m]) * scale;
  }
}

// ---------------------------------------------------------------------------
// Kernel 2: 7x7 windowed attention with reflect padding.
//   One block = one (batch, head, 8x8 pixel tile). 64 threads, 1 thread/pixel.
//   Pass A stages (k + positional-encoding) in LDS (needs VALU, sync path);
//   pass B stages the v patch with per-lane GLOBAL_LOAD_ASYNC_TO_LDS_B32.
//   Two 32-channel chunks keep smem at 25 KB.
// ---------------------------------------------------------------------------
__global__ __launch_bounds__(64) void att_kernel(
    const float* __restrict__ ws,
    const float* __restrict__ Wp, const float* __restrict__ bp,
    const float* __restrict__ rate1p,
    float* __restrict__ out)
{
  __shared__ float smem[32 * 196];           // 32 channels x 14x14 patch
  const int tid  = threadIdx.x;
  const int bh   = blockIdx.x / 36;          // b*HEAD + head
  const int tile = blockIdx.x % 36;
  const int b    = bh >> 2;
  const int head = bh & 3;
  const int ty0  = (tile / 6) * 8;
  const int tx0  = (tile % 6) * 8;
  const int ly = tid >> 3, lx = tid & 7;
  const int y = ty0 + ly, x = tx0 + lx;

  const float* q = ws;
  const float* k = ws + (size_t)QSIZE_;
  const float* v = ws + (size_t)(2) * QSIZE_;
  const size_t cbase = (size_t)b * PLANE_ + (size_t)(head * HD_) * N_;
  const float sloc = 2.0f / 47.0f;           // linspace(-1,1,48) step

  float att[49];
  #pragma unroll
  for (int i = 0; i < 49; ++i) att[i] = 0.f;

  // ---- pass A: att[k] = sum_d q_d * (k + pe)[window] ----
  for (int ch = 0; ch < 2; ++ch) {
    for (int idx = tid; idx < 32 * 196; idx += 64) {
      int dd = idx / 196, pix = idx % 196;
      int d  = ch * 32 + dd;
      int py = pix / 14, px = pix % 14;
      int ry = refl(ty0 - 3 + py);
      int rx = refl(tx0 - 3 + px);
      float pe = Wp[2*d] * (-1.0f + rx * sloc)
               + Wp[2*d+1] * (-1.0f + ry * sloc) + bp[d];
      smem[idx] = k[cbase + (size_t)d * N_ + ry * W_ + rx] + pe;
    }
    __syncthreads();
    const float* qp = q + cbase + (size_t)(ch * 32) * N_ + y * W_ + x;
    for (int dd = 0; dd < 32; ++dd) {
      float qd = qp[(size_t)dd * N_];
      const float* sr = smem + dd * 196 + ly * 14 + lx;
      #pragma unroll
      for (int ky = 0; ky < 7; ++ky)
        #pragma unroll
        for (int kx = 0; kx < 7; ++kx)
          att[ky*7 + kx] = fmaf(qd, sr[ky*14 + kx], att[ky*7 + kx]);
    }
    __syncthreads();
  }

  // ---- softmax over 49 (registers only), fold in 1/sum and rate1 ----
  float mx = att[0];
  #pragma unroll
  for (int i = 1; i < 49; ++i) mx = fmaxf(mx, att[i]);
  float sum = 0.f;
  #pragma unroll
  for (int i = 0; i < 49; ++i) { att[i] = __expf(att[i] - mx); sum += att[i]; }
  const float inv = rate1p[0] / sum;
  #pragma unroll
  for (int i = 0; i < 49; ++i) att[i] *= inv;

  // ---- pass B: out_d = sum_k att[k] * v[window]; v patch via async->LDS ----
  for (int ch = 0; ch < 2; ++ch) {
    for (int idx = tid; idx < 32 * 196; idx += 64) {
      int dd = idx / 196, pix = idx % 196;
      int d  = ch * 32 + dd;
      int py = pix / 14, px = pix % 14;
      int ry = refl(ty0 - 3 + py);
      int rx = refl(tx0 - 3 + px);
      unsigned laddr = (unsigned)(size_t)&smem[idx];
      async_g2l_b32(laddr, v + cbase + (size_t)d * N_ + ry * W_ + rx);
    }
    wait_async0();
    __syncthreads();
    for (int dd = 0; dd < 32; ++dd) {
      const float* sr = smem + dd * 196 + ly * 14 + lx;
      float o = 0.f;
      #pragma unroll
      for (int ky = 0; ky < 7; ++ky)
        #pragma unroll
        for (int kx = 0; kx < 7; ++kx)
          o = fmaf(att[ky*7 + kx], sr[ky*14 + kx], o);
      out[cbase + (size_t)(ch * 32 + dd) * N_ + y * W_ + x] = o;
    }
    __syncthreads();
  }
}

// ---------------------------------------------------------------------------
// Kernel 3: collapsed dynamic-conv branch.
//   All 9 shift replicas share the same depthwise filters and the shift
//   softmax sums to 1 => out_conv = dw3x3(relu(dw3x3(x))) (zero padding).
//   The Wfc GEMM + softmax are dead work and never executed.
//   One block = one (batch*channel, 16x16 tile). Accumulates += rate2*val.
// ---------------------------------------------------------------------------
__global__ __launch_bounds__(256) void conv_kernel(
    const float* __restrict__ x,
    const float* __restrict__ w1, const float* __restrict__ w2,
    const float* __restrict__ rate2p,
    float* __restrict__ out)
{
  __shared__ float sx[20 * 20];
  __shared__ float sh[18 * 18];
  const int tid  = threadIdx.x;
  const int bc   = blockIdx.x / 9;           // b*256 + c plane index
  const int tile = blockIdx.x % 9;
  const int ty0  = (tile / 3) * 16;
  const int tx0  = (tile % 3) * 16;
  const int c    = bc & 255;
  const float* xp = x + (size_t)bc * N_;

  float a1[9], a2[9];
  #pragma unroll
  for (int i = 0; i < 9; ++i) { a1[i] = w1[c*9 + i]; a2[i] = w2[c*9 + i]; }

  for (int idx = tid; idx < 400; idx += 256) {
    int py = idx / 20, px = idx % 20;
    int gy = ty0 - 2 + py, gx = tx0 - 2 + px;
    sx[idx] = (gy >= 0 && gy < H_ && gx >= 0 && gx < W_) ? xp[gy * W_ + gx] : 0.f;
  }
  __syncthreads();

  for (int idx = tid; idx < 324; idx += 256) {
    int hy = idx / 18, hx = idx % 18;
    int gy = ty0 - 1 + hy, gx = tx0 - 1 + hx;
    float acc = 0.f;
    #pragma unroll
    for (int i = 0; i < 3; ++i)
      #pragma unroll
      for (int j = 0; j < 3; ++j)
        acc = fmaf(a1[i*3 + j], sx[(hy + i) * 20 + hx + j], acc);
    // hid only exists inside the image; conv2's zero padding supplies 0 outside
    sh[idx] = (gy >= 0 && gy < H_ && gx >= 0 && gx < W_) ? fmaxf(acc, 0.f) : 0.f;
  }
  __syncthreads();

  const int ly = tid >> 4, lx = tid & 15;
  float acc = 0.f;
  #pragma unroll
  for (int i = 0; i < 3; ++i)
    #pragma unroll
    for (int j = 0; j < 3; ++j)
      acc = fmaf(a2[i*3 + j], sh[(ly + i) * 18 + lx + j], acc);

  out[(size_t)bc * N_ + (ty0 + ly) * W_ + tx0 + lx] += rate2p[0] * acc;
}

// ---------------------------------------------------------------------------
extern "C" void kernel_launch(void* const* d_in, const int* in_sizes, int n_in,
                              void* d_out, int out_size, void* d_ws, size_t ws_size,
                              hipStream_t stream) {
  const float* x  = (const float*)d_in[0];
  const float* Wq = (const float*)d_in[1];
  const float* bq = (const float*)d_in[2];
  const float* Wk = (const float*)d_in[3];
  const float* bk = (const float*)d_in[4];
  const float* Wv = (const float*)d_in[5];
  const float* bv = (const float*)d_in[6];
  const float* Wp = (const float*)d_in[7];
  const float* bp = (const float*)d_in[8];
  // d_in[9] = Wfc: mathematically dead (shift softmax sums to 1 over identical
  // depthwise replicas) -- never read.
  const float* w1 = (const float*)d_in[10];
  const float* w2 = (const float*)d_in[11];
  const float* r1 = (const float*)d_in[12];
  const float* r2 = (const float*)d_in[13];
  float* out = (float*)d_out;
  float* ws  = (float*)d_ws;    // q | k | v, QSIZE_ floats each (13.5 MB total)

  dim3 gq(16 * 72, 3);          // 16 m-tiles x 72 n-groups; y selects q/k/v
  qkv_wmma_kernel<<<gq, 128, 0, stream>>>(x, Wq, bq, Wk, bk, Wv, bv, ws);
  att_kernel<<<B_ * HEAD_ * 36, 64, 0, stream>>>(ws, Wp, bp, r1, out);
  conv_kernel<<<B_ * C_ * 9, 256, 0, stream>>>(x, w1, w2, r2, out);
}